// GNNNextActivity_45767171506442
// MI455X (gfx1250) — compile-verified
//
#include <hip/hip_runtime.h>

typedef __attribute__((ext_vector_type(2))) float v2f;
typedef __attribute__((ext_vector_type(8))) float v8f;

#define NN     100000   // nodes
#define NE     1600000  // edges
#define INC    128
#define HIDC   256
#define OUTC   128
#define NG     2048     // graphs
#define NB     ((NN + 255) / 256)   // scan blocks

// ---------------------------------------------------------------------------
// Zero fill (graph-capture safe)
// ---------------------------------------------------------------------------
__global__ void fill_zero(float* __restrict__ p, size_t n) {
    size_t i = ((size_t)blockIdx.x * blockDim.x + threadIdx.x) * 4;
    if (i + 4 <= n) {
        *reinterpret_cast<float4*>(p + i) = make_float4(0.f, 0.f, 0.f, 0.f);
    } else {
        for (; i < n; ++i) p[i] = 0.f;
    }
}

// ---------------------------------------------------------------------------
// CSR construction (built once, reused by both SAGE layers)
// ---------------------------------------------------------------------------
__global__ void edge_hist(const int* __restrict__ dst, int* __restrict__ deg, int nEdges) {
    int e = blockIdx.x * blockDim.x + threadIdx.x;
    if (e < nEdges) atomicAdd(&deg[dst[e]], 1);
}

__global__ void block_reduce_deg(const int* __restrict__ deg, int* __restrict__ bsum, int n) {
    __shared__ int sh[256];
    int i = blockIdx.x * 256 + threadIdx.x;
    sh[threadIdx.x] = (i < n) ? deg[i] : 0;
    __syncthreads();
    for (int s = 128; s > 0; s >>= 1) {
        if (threadIdx.x < s) sh[threadIdx.x] += sh[threadIdx.x + s];
        __syncthreads();
    }
    if (threadIdx.x == 0) bsum[blockIdx.x] = sh[0];
}

__global__ void scan_bsums(const int* __restrict__ bsum, int* __restrict__ boff, int nb) {
    if (blockIdx.x == 0 && threadIdx.x == 0) {
        int acc = 0;
        for (int b = 0; b < nb; ++b) { boff[b] = acc; acc += bsum[b]; }
    }
}

// Per-block exclusive scan of degrees -> node offsets; also emits cursor copy
// (consumed by fill_csr) and float degree (consumed by the GEMM's A1 row scaling).
__global__ void node_offsets(const int* __restrict__ deg, const int* __restrict__ boff,
                             int* __restrict__ off, int* __restrict__ cursor,
                             float* __restrict__ cnt, int n) {
    __shared__ int sh[256];
    int i = blockIdx.x * 256 + threadIdx.x;
    int d = (i < n) ? deg[i] : 0;
    sh[threadIdx.x] = d;
    __syncthreads();
    for (int s = 1; s < 256; s <<= 1) {           // Hillis-Steele inclusive scan
        int v = (threadIdx.x >= s) ? sh[threadIdx.x - s] : 0;
        __syncthreads();
        sh[threadIdx.x] += v;
        __syncthreads();
    }
    if (i < n) {
        int excl = sh[threadIdx.x] - d + boff[blockIdx.x];
        off[i]    = excl;
        cursor[i] = excl;
        cnt[i]    = (float)d;
    }
}

__global__ void fill_csr(const int* __restrict__ src, const int* __restrict__ dst,
                         int* __restrict__ cursor, int* __restrict__ esrc, int nEdges) {
    int e = blockIdx.x * blockDim.x + threadIdx.x;
    if (e < nEdges) {
        int p = atomicAdd(&cursor[dst[e]], 1);
        esrc[p] = src[e];
    }
}

// ---------------------------------------------------------------------------
// Gather aggregation: agg[i] = sum over in-edges of X[src]. No atomics.
// C >= 128 -> the 4-channel groups of one node cover whole waves, so the node
// index is wave-uniform: readfirstlane turns CSR metadata into scalar loads,
// and each edge's feature row becomes one coalesced 512B/1KB wave read.
// ---------------------------------------------------------------------------
__global__ void gather_agg(const float* __restrict__ X, const int* __restrict__ off,
                           const int* __restrict__ deg, const int* __restrict__ esrc,
                           float* __restrict__ agg, int nNodes, int C) {
    const int c4 = C >> 2;
    size_t tid = (size_t)blockIdx.x * blockDim.x + threadIdx.x;
    if (tid >= (size_t)nNodes * c4) return;
    int i = __builtin_amdgcn_readfirstlane((int)(tid / c4));   // wave-uniform
    int g = (int)(tid % c4);
    int s0 = __builtin_amdgcn_readfirstlane(off[i]);
    int d  = __builtin_amdgcn_readfirstlane(deg[i]);
    float4 acc = make_float4(0.f, 0.f, 0.f, 0.f);
    for (int e = 0; e < d; ++e) {
        int s = __builtin_amdgcn_readfirstlane(esrc[s0 + e]);  // scalar load
        float4 v = *reinterpret_cast<const float4*>(X + (size_t)s * C + g * 4);
        acc.x += v.x; acc.y += v.y; acc.z += v.z; acc.w += v.w;
    }
    *reinterpret_cast<float4*>(agg + (size_t)i * C + g * 4) = acc;
}

// ---------------------------------------------------------------------------
// Node -> graph pooling (small: ~25.7M atomics)
// ---------------------------------------------------------------------------
__global__ void pool_nodes(const float* __restrict__ H, const int* __restrict__ batch,
                           float* __restrict__ psum, float* __restrict__ gcnt,
                           int nNodes, int C) {
    const int c4 = C >> 2;
    size_t tid = (size_t)blockIdx.x * blockDim.x + threadIdx.x;
    if (tid >= (size_t)nNodes * c4) return;
    int i = (int)(tid / c4);
    int g = (int)(tid % c4);
    int b = batch[i];
    float4 v = *reinterpret_cast<const float4*>(H + (size_t)i * C + g * 4);
    float* out = psum + (size_t)b * C + g * 4;
    atomicAdd(out + 0, v.x);
    atomicAdd(out + 1, v.y);
    atomicAdd(out + 2, v.z);
    atomicAdd(out + 3, v.w);
    if (g == 0) atomicAdd(gcnt + b, 1.0f);
}

// ---------------------------------------------------------------------------
// Fused WMMA GEMM:
//   C[M,N] = act( (A1 .* 1/max(cnt,1)) @ B1  [+ A2 @ B2]  + bias )
// One wave = one 16x16 tile; K stepped by 4 via V_WMMA_F32_16X16X4_F32.
// A 16x4 : lanes 0-15 -> row=lane, {K,K+1}; lanes 16-31 -> row=lane-16, {K+2,K+3}
// B 4x16 : col = lane%16; b.x = K(+2 hi half), b.y = K+1(+3)
// C 16x16: vgpr v -> row = v + 8*(lane>=16), col = lane%16
// ---------------------------------------------------------------------------
template <bool HAS_A2, bool RELU>
__global__ __launch_bounds__(256) void wmma_gemm(
        const float* __restrict__ A1, const float* __restrict__ cnt,
        const float* __restrict__ B1,
        const float* __restrict__ A2, const float* __restrict__ B2,
        const float* __restrict__ bias,
        float* __restrict__ Cm,
        int M, int N, int K) {
    const int lane = threadIdx.x & 31;
    const int wave = threadIdx.x >> 5;
    const int half = lane >> 4;
    const int l16  = lane & 15;

    const int tilesN   = N >> 4;
    const int totTiles = (M >> 4) * tilesN;
    const int tile     = blockIdx.x * 8 + wave;
    if (tile >= totTiles) return;                 // wave-uniform: EXEC stays all-1s

    const int rt = tile / tilesN;
    const int ct = tile % tilesN;
    const int rowA = rt * 16 + l16;
    const int colB = ct * 16 + l16;

    float scale = 1.0f;
    if (cnt != nullptr) scale = 1.0f / fmaxf(cnt[rowA], 1.0f);

    const float* a1p = A1 + (size_t)rowA * K;
    const float* a2p = HAS_A2 ? (A2 + (size_t)rowA * K) : nullptr;

    v8f acc = {};
    for (int k = 0; k < K; k += 4) {
        const int ka = k + half * 2;
        v2f a1, b1;
        a1.x = a1p[ka] * scale;
        a1.y = a1p[ka + 1] * scale;
        b1.x = B1[(size_t)ka * N + colB];
        b1.y = B1[(size_t)(ka + 1) * N + colB];
        acc = __builtin_amdgcn_wmma_f32_16x16x4_f32(false, a1, false, b1,
                                                    (short)0, acc, false, false);
        if (HAS_A2) {
            v2f a2, b2;
            a2.x = a2p[ka];
            a2.y = a2p[ka + 1];
            b2.x = B2[(size_t)ka * N + colB];
            b2.y = B2[(size_t)(ka + 1) * N + colB];
            acc = __builtin_amdgcn_wmma_f32_16x16x4_f32(false, a2, false, b2,
                                                        (short)0, acc, false, false);
        }
    }

    const float bv = bias[ct * 16 + l16];
#pragma unroll
    for (int i = 0; i < 8; ++i) {
        float r = acc[i] + bv;
        if (RELU) r = fmaxf(r, 0.0f);
        Cm[(size_t)(rt * 16 + half * 8 + i) * N + ct * 16 + l16] = r;
    }
}

// ---------------------------------------------------------------------------
// Launch
// ---------------------------------------------------------------------------
static inline size_t alignUp(size_t v, size_t a) { return (v + a - 1) & ~(a - 1); }

extern "C" void kernel_launch(void* const* d_in, const int* in_sizes, int n_in,
                              void* d_out, int out_size, void* d_ws, size_t ws_size,
                              hipStream_t stream) {
    (void)in_sizes; (void)n_in; (void)out_size; (void)ws_size;
    const float* x    = (const float*)d_in[0];
    const int*   ei   = (const int*)d_in[1];      // [2, NE]
    const int*   bat  = (const int*)d_in[2];
    const float* W1l  = (const float*)d_in[3];
    const float* b1   = (const float*)d_in[4];
    const float* W1r  = (const float*)d_in[5];
    const float* W2l  = (const float*)d_in[6];
    const float* b2   = (const float*)d_in[7];
    const float* W2r  = (const float*)d_in[8];
    const float* W3   = (const float*)d_in[9];
    const float* b3   = (const float*)d_in[10];
    const float* W4   = (const float*)d_in[11];
    const float* b4   = (const float*)d_in[12];
    const int* src = ei;
    const int* dst = ei + NE;

    // Workspace layout. Buffers needing zeroing come first (one contiguous span).
    char* base = (char*)d_ws;
    size_t o = 0;
    auto alloc = [&](size_t bytes) { void* p = base + o; o = alignUp(o + bytes, 256); return p; };

    int*   deg   = (int*)  alloc((size_t)NN * 4);        // zeroed
    float* psum  = (float*)alloc((size_t)NG * HIDC * 4); // zeroed
    float* gcnt  = (float*)alloc((size_t)NG * 4);        // zeroed
    const size_t zeroBytes = o;
    int*   off   = (int*)  alloc((size_t)NN * 4);
    int*   cursor= (int*)  alloc((size_t)NN * 4);
    int*   esrc  = (int*)  alloc((size_t)NE * 4);
    int*   bsum  = (int*)  alloc((size_t)NB * 4);
    int*   boff  = (int*)  alloc((size_t)NB * 4);
    float* cnt   = (float*)alloc((size_t)NN * 4);
    float* agg1  = (float*)alloc((size_t)NN * INC * 4);
    float* h1    = (float*)alloc((size_t)NN * HIDC * 4);
    float* agg2  = (float*)alloc((size_t)NN * HIDC * 4);
    float* h2    = (float*)alloc((size_t)NN * HIDC * 4);
    float* hid   = (float*)alloc((size_t)NG * HIDC * 4);

    // 0) zero accumulation buffers
    {
        size_t n = zeroBytes / 4;
        fill_zero<<<(unsigned)((n / 4 + 256) / 256), 256, 0, stream>>>((float*)base, n);
    }

    // 1) CSR build: histogram -> two-level scan -> bucket fill
    edge_hist<<<(NE + 255) / 256, 256, 0, stream>>>(dst, deg, NE);
    block_reduce_deg<<<NB, 256, 0, stream>>>(deg, bsum, NN);
    scan_bsums<<<1, 32, 0, stream>>>(bsum, boff, NB);
    node_offsets<<<NB, 256, 0, stream>>>(deg, boff, off, cursor, cnt, NN);
    fill_csr<<<(NE + 255) / 256, 256, 0, stream>>>(src, dst, cursor, esrc, NE);

    // 2) layer-1 neighbor sum by gather (no atomics)
    {
        size_t total = (size_t)NN * (INC / 4);
        gather_agg<<<(unsigned)((total + 255) / 256), 256, 0, stream>>>(
            x, off, deg, esrc, agg1, NN, INC);
    }

    // 3) h1 = relu(mean1 @ W1l + b1 + x @ W1r)
    {
        int tiles = (NN / 16) * (HIDC / 16);
        wmma_gemm<true, true><<<(tiles + 7) / 8, 256, 0, stream>>>(
            agg1, cnt, W1l, x, W1r, b1, h1, NN, HIDC, INC);
    }

    // 4) layer-2 neighbor sum by gather (CSR reused)
    {
        size_t total = (size_t)NN * (HIDC / 4);
        gather_agg<<<(unsigned)((total + 255) / 256), 256, 0, stream>>>(
            h1, off, deg, esrc, agg2, NN, HIDC);
    }

    // 5) h2 = relu(mean2 @ W2l + b2 + h1 @ W2r)
    {
        int tiles = (NN / 16) * (HIDC / 16);
        wmma_gemm<true, true><<<(tiles + 7) / 8, 256, 0, stream>>>(
            agg2, cnt, W2l, h1, W2r, b2, h2, NN, HIDC, HIDC);
    }

    // 6) global mean pool (sums + per-graph node counts)
    {
        size_t total = (size_t)NN * (HIDC / 4);
        pool_nodes<<<(unsigned)((total + 255) / 256), 256, 0, stream>>>(
            h2, bat, psum, gcnt, NN, HIDC);
    }

    // 7) hidden = relu(pooled_mean @ W3 + b3)
    {
        int tiles = (NG / 16) * (HIDC / 16);
        wmma_gemm<false, true><<<(tiles + 7) / 8, 256, 0, stream>>>(
            psum, gcnt, W3, nullptr, nullptr, b3, hid, NG, HIDC, HIDC);
    }

    // 8) out = hidden @ W4 + b4
    {
        int tiles = (NG / 16) * (OUTC / 16);
        wmma_gemm<false, false><<<(tiles + 7) / 8, 256, 0, stream>>>(
            hid, nullptr, W4, nullptr, nullptr, b4, (float*)d_out, NG, OUTC, HIDC);
    }
}